// Model_9285719294186
// MI455X (gfx1250) — compile-verified
//
#include <hip/hip_runtime.h>
#include <hip/hip_bf16.h>
#include <math.h>

// ---------------------------------------------------------------------------
// Model constants
// ---------------------------------------------------------------------------
#define BATCH 16
#define NPTS  2048
#define FEAT  352
#define EMB   128
#define CODE_DIM 1024

typedef float v2f __attribute__((ext_vector_type(2)));
typedef float v8f __attribute__((ext_vector_type(8)));

__device__ __forceinline__ int imin(int a, int b) { return a < b ? a : b; }

// ---------------------------------------------------------------------------
// Generic fp32 WMMA GEMM:  C[M,N] = A[M,K] @ W[N,K]^T + bias[N], optional ReLU
//
// Block = 256 threads = 8 waves. A block owns 8 M-tiles (128 rows) x one
// 64-column N-strip. Per 64-wide K chunk:
//   * all threads cooperatively stage A (128x64) and W-strip (64x64) into LDS,
//     coalesced along K, zero-padded past K (kills the K-tail path entirely),
//   * prefetch the next chunk (global_prefetch_b8),
//   * each wave runs 16 k-steps x 4 WMMAs fed from ds_load (stride 68 padding
//     => conflict-free banks).
// M must be a multiple of 16 (true everywhere in this model). Masked columns
// use clamped in-bounds rows; their garbage only feeds never-stored columns.
// ---------------------------------------------------------------------------
#define KC   64
#define LSTR 68   // padded LDS row stride (floats)

__global__ __launch_bounds__(256)
void wmma_gemm_f32(const float* __restrict__ A, const float* __restrict__ W,
                   const float* __restrict__ bias, float* __restrict__ C,
                   int M, int N, int K, int act) {
    __shared__ float As[128 * LSTR];   // 34816 B
    __shared__ float Ws[64 * LSTR];    // 17408 B

    const int tid    = (int)threadIdx.x;
    const int waveId = tid >> 5;
    const int lane   = tid & 31;
    const int half   = lane >> 4;
    const int l16    = lane & 15;

    const int mtile  = M >> 4;
    const int nstrip = (N + 63) >> 6;
    const int ts     = (int)blockIdx.x % nstrip;   // N-strip of this block
    const int mb     = (int)blockIdx.x / nstrip;   // M-block (8 tiles)
    const int tm     = mb * 8 + waveId;
    const bool active = (tm < mtile);

    v8f acc0 = {}, acc1 = {}, acc2 = {}, acc3 = {};

    for (int kc0 = 0; kc0 < K; kc0 += KC) {
        __syncthreads();   // previous chunk's readers are done

        // ---- stage A chunk: rows mb*128..+127, k kc0..+63 (coalesced in K)
        for (int e = tid; e < 128 * KC; e += 256) {
            int r = e >> 6, kk = e & 63;
            int grow = imin(mb * 128 + r, M - 1);
            int gk   = kc0 + kk;
            float v = A[(size_t)grow * K + imin(gk, K - 1)];
            As[r * LSTR + kk] = (gk < K) ? v : 0.0f;
        }
        // ---- stage W strip chunk: cols ts*64..+63, k kc0..+63
        for (int e = tid; e < 64 * KC; e += 256) {
            int c = e >> 6, kk = e & 63;
            int gcol = imin(ts * 64 + c, N - 1);
            int gk   = kc0 + kk;
            float v = W[(size_t)gcol * K + imin(gk, K - 1)];
            Ws[c * LSTR + kk] = (gk < K) ? v : 0.0f;
        }
        // ---- prefetch next chunk while this one is consumed
        if (kc0 + KC < K) {
            int pk = kc0 + KC;
            {   // W: 64 cols x 2 cachelines
                int c  = tid >> 2;
                int kk = (tid & 3) * 16;
                const float* p = W + (size_t)imin(ts * 64 + c, N - 1) * K
                                   + imin(pk + kk, K - 1);
                __builtin_prefetch(p, 0, 3);
            }
            {   // A: 128 rows x 2 cachelines
                int r  = tid >> 1;
                int kk = (tid & 1) * 32;
                const float* p = A + (size_t)imin(mb * 128 + r, M - 1) * K
                                   + imin(pk + kk, K - 1);
                __builtin_prefetch(p, 0, 3);
            }
        }
        __syncthreads();

        // ---- compute: 16 k-steps x 4 WMMAs from LDS
        if (active) {
            const float* Abase = &As[(waveId * 16 + l16) * LSTR];
            const float* W0 = &Ws[(l16     ) * LSTR];
            const float* W1 = &Ws[(l16 + 16) * LSTR];
            const float* W2 = &Ws[(l16 + 32) * LSTR];
            const float* W3 = &Ws[(l16 + 48) * LSTR];
            #pragma unroll 4
            for (int kk = 0; kk < KC; kk += 4) {
                v2f a = *(const v2f*)(Abase + kk + 2 * half);
                int i0 = kk + half, i1 = kk + 2 + half;
                v2f b0, b1, b2, b3;
                b0.x = W0[i0];  b0.y = W0[i1];
                b1.x = W1[i0];  b1.y = W1[i1];
                b2.x = W2[i0];  b2.y = W2[i1];
                b3.x = W3[i0];  b3.y = W3[i1];
                acc0 = __builtin_amdgcn_wmma_f32_16x16x4_f32(false, a, false, b0, (short)0, acc0, false, false);
                acc1 = __builtin_amdgcn_wmma_f32_16x16x4_f32(false, a, false, b1, (short)0, acc1, false, false);
                acc2 = __builtin_amdgcn_wmma_f32_16x16x4_f32(false, a, false, b2, (short)0, acc2, false, false);
                acc3 = __builtin_amdgcn_wmma_f32_16x16x4_f32(false, a, false, b3, (short)0, acc3, false, false);
            }
        }
    }

    if (!active) return;

    // C/D layout: VGPR r holds M = tm*16 + r + 8*half, column = lane%16 of tile
    size_t crow = (size_t)(tm * 16 + 8 * half) * N;
    int ncol0 = ts * 64 + l16;
    #pragma unroll
    for (int t = 0; t < 4; ++t) {
        int nc = ncol0 + t * 16;
        v8f accv = (t == 0) ? acc0 : (t == 1) ? acc1 : (t == 2) ? acc2 : acc3;
        if (nc < N) {
            float bv = bias ? bias[nc] : 0.0f;
            #pragma unroll
            for (int r = 0; r < 8; ++r) {
                float v = accv[r] + bv;
                if (act == 1) v = fmaxf(v, 0.0f);
                C[crow + (size_t)r * N + nc] = v;
            }
        }
    }
}

// ---------------------------------------------------------------------------
// nearest_idx[b,n] = argmin_m || pc[b,n] - mirror_x(pc[b,m]) ||
// ---------------------------------------------------------------------------
__global__ __launch_bounds__(256)
void nearest_kernel(const float* __restrict__ pc, int* __restrict__ nidx) {
    int b = blockIdx.x;
    int n = blockIdx.y * 256 + threadIdx.x;
    __shared__ float sx[NPTS], sy[NPTS], sz[NPTS];
    for (int m = threadIdx.x; m < NPTS; m += 256) {
        const float* p = pc + ((size_t)b * NPTS + m) * 3;
        sx[m] = p[0]; sy[m] = p[1]; sz[m] = p[2];
    }
    __syncthreads();
    const float* p = pc + ((size_t)b * NPTS + n) * 3;
    float px = p[0], py = p[1], pz = p[2];
    float best = 3.4e38f; int bi = 0;
    for (int m = 0; m < NPTS; ++m) {
        float dx = px + sx[m];   // px - (-sx[m])
        float dy = py - sy[m];
        float dz = pz - sz[m];
        float d2 = dx*dx + dy*dy + dz*dz;
        if (d2 < best) { best = d2; bi = m; }
    }
    nidx[b * NPTS + n] = bi;
}

// ---------------------------------------------------------------------------
// Per point: z = sigmoid(emb_raw[128]); emb_out = normalize(emb_raw[0:128])
// ---------------------------------------------------------------------------
__global__ __launch_bounds__(128)
void z_emb_kernel(const float* __restrict__ embRaw, float* __restrict__ zOut,
                  float* __restrict__ embOut, float* __restrict__ zWs) {
    int bn = blockIdx.x;             // 0..B*N-1
    int c  = threadIdx.x;            // 0..127
    const float* row = embRaw + (size_t)bn * 129;
    float v = row[c];
    __shared__ float red[128];
    __shared__ float denom;
    red[c] = v * v; __syncthreads();
    for (int s = 64; s > 0; s >>= 1) { if (c < s) red[c] += red[c + s]; __syncthreads(); }
    if (c == 0) {
        denom = fmaxf(sqrtf(red[0]), 1e-12f);
        float zz = 1.0f / (1.0f + expf(-row[128]));
        zOut[bn] = zz; zWs[bn] = zz;
    }
    __syncthreads();
    embOut[(size_t)bn * EMB + c] = v / denom;
}

// ---------------------------------------------------------------------------
// loss_sym accumulation: sum_c |emb_raw[b,nidx,c] - emb_raw[b,n,c]| (129 dims)
// ---------------------------------------------------------------------------
__global__ __launch_bounds__(128)
void sym_kernel(const float* __restrict__ embRaw, const int* __restrict__ nidx,
                float* __restrict__ accum) {
    int bn = blockIdx.x;
    int b  = bn >> 11;
    int c  = threadIdx.x;
    int j  = nidx[bn];
    const float* ra = embRaw + (size_t)bn * 129;
    const float* rb = embRaw + ((size_t)b * NPTS + j) * 129;
    float s = fabsf(rb[c] - ra[c]);
    if (c == 0) s += fabsf(rb[128] - ra[128]);
    __shared__ float red[128];
    red[c] = s; __syncthreads();
    for (int st = 64; st > 0; st >>= 1) { if (c < st) red[c] += red[c + st]; __syncthreads(); }
    if (c == 0) atomicAdd(accum, red[0]);
}

// ---------------------------------------------------------------------------
// Discriminator: forwards on z (fake), real_z (real), z_interp (+ manual
// backward for gradient-penalty norm). One block per batch element.
// dims: 2048 -> 512 -> 256 -> 1, ReLU between layers.
// ---------------------------------------------------------------------------
__global__ __launch_bounds__(256)
void disc_kernel(const float* __restrict__ z, const float* __restrict__ realz,
                 const float* __restrict__ tin,
                 const float* __restrict__ W1, const float* __restrict__ b1,
                 const float* __restrict__ W2, const float* __restrict__ b2,
                 const float* __restrict__ W3, const float* __restrict__ b3,
                 float* __restrict__ fakeV, float* __restrict__ realV,
                 float* __restrict__ gnV) {
    int b = blockIdx.x, tid = threadIdx.x;
    __shared__ float xin[NPTS];
    __shared__ float h1[512];
    __shared__ float h2[256];
    __shared__ float d1[512];
    __shared__ float red[256];
    float tb = tin[b];

    for (int pass = 0; pass < 3; ++pass) {
        for (int k = tid; k < NPTS; k += 256) {
            float zv = z[b * NPTS + k], rv = realz[b * NPTS + k];
            xin[k] = (pass == 0) ? zv : (pass == 1 ? rv : zv + tb * (rv - zv));
        }
        __syncthreads();
        for (int o = tid; o < 512; o += 256) {
            float s = b1[o];
            const float* w = W1 + (size_t)o * NPTS;
            for (int k = 0; k < NPTS; ++k) s += w[k] * xin[k];
            h1[o] = s;
        }
        __syncthreads();
        {
            int o = tid;
            float s = b2[o];
            const float* w = W2 + (size_t)o * 512;
            for (int k = 0; k < 512; ++k) s += w[k] * fmaxf(h1[k], 0.0f);
            h2[o] = s;
        }
        __syncthreads();
        red[tid] = W3[tid] * fmaxf(h2[tid], 0.0f);
        __syncthreads();
        for (int s = 128; s > 0; s >>= 1) { if (tid < s) red[tid] += red[tid + s]; __syncthreads(); }
        if (tid == 0) {
            float val = red[0] + b3[0];
            if (pass == 0) fakeV[b] = val;
            else if (pass == 1) realV[b] = val;
        }
        __syncthreads();
        if (pass == 2) {
            // d2[j] = W3[j] * relu'(h2[j])   (reuse red as d2)
            red[tid] = (h2[tid] > 0.0f) ? W3[tid] : 0.0f;
            __syncthreads();
            for (int i = tid; i < 512; i += 256) {
                float s = 0.0f;
                for (int j = 0; j < 256; ++j) s += red[j] * W2[(size_t)j * 512 + i];
                d1[i] = (h1[i] > 0.0f) ? s : 0.0f;
            }
            __syncthreads();
            float gsq = 0.0f;
            for (int k = tid; k < NPTS; k += 256) {
                float s = 0.0f;
                for (int i = 0; i < 512; ++i) s += d1[i] * W1[(size_t)i * NPTS + k];
                gsq += s * s;
            }
            __syncthreads();
            red[tid] = gsq; __syncthreads();
            for (int s = 128; s > 0; s >>= 1) { if (tid < s) red[tid] += red[tid + s]; __syncthreads(); }
            if (tid == 0) gnV[b] = sqrtf(red[0]);
        }
        __syncthreads();
    }
}

// ---------------------------------------------------------------------------
// code_pre[b, c]     = max_n relu( emb[b,n,c]) * z[b,n]
// code_pre[b, 128+c] = max_n relu(-emb[b,n,c]) * z[b,n]
// ---------------------------------------------------------------------------
__global__ __launch_bounds__(128)
void code_max_kernel(const float* __restrict__ emb, const float* __restrict__ z,
                     float* __restrict__ codePre) {
    int b = blockIdx.x, c = threadIdx.x;
    float mp = 0.0f, mn = 0.0f;
    for (int n = 0; n < NPTS; ++n) {
        float e  = emb[((size_t)(b * NPTS + n)) * EMB + c];
        float zz = z[b * NPTS + n];
        mp = fmaxf(mp, fmaxf(e, 0.0f) * zz);
        mn = fmaxf(mn, fmaxf(-e, 0.0f) * zz);
    }
    codePre[b * 256 + c]       = mp;
    codePre[b * 256 + 128 + c] = mn;
}

// ---------------------------------------------------------------------------
// lvl0 output (B,16) -> X0 rows=(b*2+p), cols f: X0 = tanh(v[b, f*2 + p])
// ---------------------------------------------------------------------------
__global__ void x0_kernel(const float* __restrict__ t3, float* __restrict__ X) {
    int i = blockIdx.x * blockDim.x + threadIdx.x;
    if (i >= BATCH * 16) return;
    int b = i >> 4, r = i & 15, f = r >> 1, p = r & 1;
    X[(size_t)(b * 2 + p) * 8 + f] = tanhf(t3[b * 16 + f * 2 + p]);
}

// ---------------------------------------------------------------------------
// Build Y rows=(b*P+p), cols 0..1031: [x channels(8) | code(1024) broadcast]
// ---------------------------------------------------------------------------
__global__ void build_y_kernel(const float* __restrict__ X, const float* __restrict__ code,
                               float* __restrict__ Y, int P) {
    size_t i = (size_t)blockIdx.x * blockDim.x + threadIdx.x;
    size_t total = (size_t)BATCH * P * 1032;
    if (i >= total) return;
    int c = (int)(i % 1032);
    size_t row = i / 1032;
    int b = (int)(row / P);
    Y[i] = (c < 8) ? X[row * 8 + c] : code[(size_t)b * CODE_DIM + (c - 8)];
}

// ---------------------------------------------------------------------------
// Xnext[(b*4P + t*P + p), o] = tanh(Out[(b*P + p), o*4 + t]) — TopNet reshape
// ---------------------------------------------------------------------------
__global__ void expand_kernel(const float* __restrict__ Out, float* __restrict__ Xn,
                              int P, int nout) {
    size_t i = (size_t)blockIdx.x * blockDim.x + threadIdx.x;
    size_t total = (size_t)BATCH * P * 4 * nout;
    if (i >= total) return;
    int o = (int)(i % nout);
    size_t row = i / nout;                    // b*4P + t*P + p
    int b = (int)(row / (4 * P));
    int rem = (int)(row % (4 * P));
    int t = rem / P, p = rem % P;
    Xn[i] = tanhf(Out[((size_t)(b * P + p)) * (nout * 4) + o * 4 + t]);
}

// ---------------------------------------------------------------------------
// One-directional chamfer: accum += sum_n sqrt(min_m ||A[b,n]-B[b,m]||^2)
// ---------------------------------------------------------------------------
__global__ __launch_bounds__(256)
void chamfer_kernel(const float* __restrict__ Ap, const float* __restrict__ Bp,
                    float* __restrict__ accum) {
    int b = blockIdx.x;
    int n = blockIdx.y * 256 + threadIdx.x;
    __shared__ float sx[NPTS], sy[NPTS], sz[NPTS];
    __shared__ float red[256];
    for (int m = threadIdx.x; m < NPTS; m += 256) {
        const float* p = Bp + ((size_t)b * NPTS + m) * 3;
        sx[m] = p[0]; sy[m] = p[1]; sz[m] = p[2];
    }
    __syncthreads();
    const float* p = Ap + ((size_t)b * NPTS + n) * 3;
    float px = p[0], py = p[1], pz = p[2];
    float best = 3.4e38f;
    for (int m = 0; m < NPTS; ++m) {
        float dx = px - sx[m], dy = py - sy[m], dz = pz - sz[m];
        best = fminf(best, dx*dx + dy*dy + dz*dz);
    }
    red[threadIdx.x] = sqrtf(fmaxf(best, 0.0f));
    __syncthreads();
    for (int s = 128; s > 0; s >>= 1) {
        if (threadIdx.x < s) red[threadIdx.x] += red[threadIdx.x + s];
        __syncthreads();
    }
    if (threadIdx.x == 0) atomicAdd(accum, red[0]);
}

// ---------------------------------------------------------------------------
// Final scalar losses
// ---------------------------------------------------------------------------
__global__ void finalize_kernel(const float* __restrict__ fakeV, const float* __restrict__ realV,
                                const float* __restrict__ gnV, const float* __restrict__ symSum,
                                const float* __restrict__ reconSum, float* __restrict__ outLoss) {
    if (threadIdx.x != 0 || blockIdx.x != 0) return;
    float lg = 0.0f, lp = 0.0f;
    for (int b = 0; b < BATCH; ++b) {
        lg += fakeV[b] - realV[b];
        float e = fmaxf(gnV[b] - 1.0f, 0.0f);
        lp += e * e;
    }
    const float invBN = 1.0f / ((float)BATCH * (float)NPTS);
    outLoss[0] = lg / (float)BATCH;
    outLoss[1] = lp / (float)BATCH;
    outLoss[2] = reconSum[0] * invBN;   // RECON_F = 1
    outLoss[3] = symSum[0] * invBN;     // SYM_F = 1
}

// ---------------------------------------------------------------------------
// Host orchestration
// ---------------------------------------------------------------------------
extern "C" void kernel_launch(void* const* d_in, const int* in_sizes, int n_in,
                              void* d_out, int out_size, void* d_ws, size_t ws_size,
                              hipStream_t stream) {
    (void)in_sizes; (void)n_in; (void)out_size; (void)ws_size;
    const float* pc    = (const float*)d_in[0];
    const float* pcf   = (const float*)d_in[1];
    const float* realz = (const float*)d_in[2];
    const float* tin   = (const float*)d_in[3];
    const float *encW[4], *encB[4];
    for (int i = 0; i < 4; ++i) { encW[i] = (const float*)d_in[4 + i]; encB[i] = (const float*)d_in[8 + i]; }
    const float *dW[3], *dB[3];
    for (int i = 0; i < 3; ++i) { dW[i] = (const float*)d_in[12 + i]; dB[i] = (const float*)d_in[15 + i]; }
    const float* fcW = (const float*)d_in[18];
    const float* fcB = (const float*)d_in[19];
    const float *l0W[3], *l0B[3];
    for (int i = 0; i < 3; ++i) { l0W[i] = (const float*)d_in[20 + i]; l0B[i] = (const float*)d_in[23 + i]; }
    const float *cW[5][5], *cB[5][5];
    for (int li = 0; li < 5; ++li)
        for (int j = 0; j < 5; ++j) {
            cW[li][j] = (const float*)d_in[26 + li * 5 + j];
            cB[li][j] = (const float*)d_in[51 + li * 5 + j];
        }

    // workspace carve-up (floats, 16-float aligned)
    float* ws = (float*)d_ws;
    size_t off = 0;
    auto alloc = [&](size_t nflt) { float* p = ws + off; off += (nflt + 15) & ~(size_t)15; return p; };
    float* embRaw  = alloc((size_t)BATCH * NPTS * 129);
    float* hA      = alloc((size_t)BATCH * NPTS * 128);
    float* hB      = alloc((size_t)BATCH * NPTS * 128);
    float* zws     = alloc((size_t)BATCH * NPTS);
    float* codePre = alloc(BATCH * 256);
    float* code    = alloc(BATCH * CODE_DIM);
    float* t1      = alloc(BATCH * 256);
    float* t2      = alloc(BATCH * 64);
    float* t3      = alloc(BATCH * 16);
    float* fakeV   = alloc(16);
    float* realV   = alloc(16);
    float* gnV     = alloc(16);
    float* acc     = alloc(16);                 // acc[0]=sym_sum, acc[1]=recon_sum
    float* X       = alloc((size_t)8192 * 8);
    float* rec     = alloc((size_t)BATCH * NPTS * 3);
    float* buf0    = alloc((size_t)8192 * 1032);
    float* buf1    = alloc((size_t)8192 * 1032);
    int*   nidx    = (int*)alloc((size_t)BATCH * NPTS);

    float* outZ    = (float*)d_out;
    float* outEmb  = outZ + (size_t)BATCH * NPTS;
    float* outLoss = outEmb + (size_t)BATCH * NPTS * EMB;

    hipMemsetAsync(acc, 0, 2 * sizeof(float), stream);

    auto gemm = [&](const float* A, const float* W, const float* bias, float* C,
                    int M, int N, int K, int act) {
        int mt = M >> 4, ns = (N + 63) >> 6;
        int mblocks = (mt + 7) / 8;
        wmma_gemm_f32<<<mblocks * ns, 256, 0, stream>>>(A, W, bias, C, M, N, K, act);
    };

    // symmetry nearest neighbour
    nearest_kernel<<<dim3(BATCH, NPTS / 256), 256, 0, stream>>>(pc, nidx);

    // encoder MLP (WMMA)
    gemm(pcf, encW[0], encB[0], hA, BATCH * NPTS, 128, FEAT, 1);
    gemm(hA,  encW[1], encB[1], hB, BATCH * NPTS, 128, 128, 1);
    gemm(hB,  encW[2], encB[2], hA, BATCH * NPTS, 128, 128, 1);
    gemm(hA,  encW[3], encB[3], embRaw, BATCH * NPTS, 129, 128, 0);

    z_emb_kernel<<<BATCH * NPTS, 128, 0, stream>>>(embRaw, outZ, outEmb, zws);
    sym_kernel<<<BATCH * NPTS, 128, 0, stream>>>(embRaw, nidx, acc + 0);
    disc_kernel<<<BATCH, 256, 0, stream>>>(zws, realz, tin, dW[0], dB[0], dW[1], dB[1],
                                           dW[2], dB[2], fakeV, realV, gnV);

    // latent code (WMMA)
    code_max_kernel<<<BATCH, 128, 0, stream>>>(outEmb, zws, codePre);
    gemm(codePre, fcW, fcB, code, BATCH, CODE_DIM, 256, 0);

    // TopNet level 0 MLP (WMMA)
    gemm(code, l0W[0], l0B[0], t1, BATCH, 256, 1024, 1);
    gemm(t1,   l0W[1], l0B[1], t2, BATCH, 64,  256,  1);
    gemm(t2,   l0W[2], l0B[2], t3, BATCH, 16,  64,   0);
    x0_kernel<<<1, 256, 0, stream>>>(t3, X);

    // TopNet conv levels 1..5 (all WMMA)
    int Ps[5] = {2, 8, 32, 128, 512};
    for (int li = 0; li < 5; ++li) {
        int P = Ps[li], rows = BATCH * P, nout = (li < 4) ? 8 : 3;
        size_t ytot = (size_t)rows * 1032;
        build_y_kernel<<<(unsigned)((ytot + 255) / 256), 256, 0, stream>>>(X, code, buf0, P);
        int dims[6] = {1032, 1032, 516, 258, 129, nout * 4};
        float* a = buf0; float* bb = buf1;
        for (int j = 0; j < 5; ++j) {
            gemm(a, cW[li][j], cB[li][j], bb, rows, dims[j + 1], dims[j], (j < 4) ? 1 : 0);
            float* tmp = a; a = bb; bb = tmp;
        }
        size_t etot = (size_t)rows * 4 * nout;
        if (li < 4)
            expand_kernel<<<(unsigned)((etot + 255) / 256), 256, 0, stream>>>(a, X, P, 8);
        else
            expand_kernel<<<(unsigned)((etot + 255) / 256), 256, 0, stream>>>(a, rec, P, 3);
    }

    // chamfer (both directions) + final losses
    chamfer_kernel<<<dim3(BATCH, NPTS / 256), 256, 0, stream>>>(pc, rec, acc + 1);
    chamfer_kernel<<<dim3(BATCH, NPTS / 256), 256, 0, stream>>>(rec, pc, acc + 1);
    finalize_kernel<<<1, 1, 0, stream>>>(fakeV, realV, gnV, acc + 0, acc + 1, outLoss);
}